// NodeBlock_24807731101812
// MI455X (gfx1250) — compile-verified
//
#include <hip/hip_runtime.h>

typedef float v2f __attribute__((ext_vector_type(2)));
typedef float v8f __attribute__((ext_vector_type(8)));

#define E_DIM     32
#define NODE_DIM  128
#define IN_DIM    224   // 2*32 + 128 + 32
#define HID       256
#define MTILE     64
#define LDS_STRIDE 260  // padded: conflict-free b64 A-frag reads (bank stride 4)

// ---------------- kernel 1: zero scratch aggregates ----------------
__global__ void zero_ws_kernel(float* __restrict__ p, int n) {
    int i = blockIdx.x * blockDim.x + threadIdx.x;
    int stride = gridDim.x * blockDim.x;
    for (; i < n; i += stride) p[i] = 0.0f;
}

// ---------------- kernel 2: edge scatter (both segment sums fused) ----------------
__global__ __launch_bounds__(256) void edge_scatter_kernel(
    const float* __restrict__ edge_attrs,
    const int*   __restrict__ senders,
    const int*   __restrict__ receivers,
    float* __restrict__ rec_agg,
    float* __restrict__ send_agg,
    int n_edges)
{
    int lane = threadIdx.x & 31;
    int e = blockIdx.x * 8 + (threadIdx.x >> 5);   // one wave32 per edge
    if (e >= n_edges) return;
    float v = edge_attrs[e * E_DIM + lane];
    int r = receivers[e];
    int s = senders[e];
    atomicAdd(&rec_agg[r * E_DIM + lane], v);
    atomicAdd(&send_agg[s * E_DIM + lane], v);
}

// ---------------- kernel 3: fused concat + 2-layer MLP with fp32 WMMA ----------------
__global__ __launch_bounds__(256) void node_mlp_kernel(
    const float* __restrict__ rec_agg,
    const float* __restrict__ send_agg,
    const float* __restrict__ node_attrs,
    const float* __restrict__ gattr,
    const float* __restrict__ W1, const float* __restrict__ b1,
    const float* __restrict__ W2, const float* __restrict__ b2,
    float* __restrict__ out, int n_nodes)
{
    __shared__ float smem[MTILE * LDS_STRIDE];

    const int tid   = threadIdx.x;
    const int wave  = tid >> 5;        // 0..7
    const int lane  = tid & 31;
    const int ln    = lane & 15;       // N (or M) index within subtile
    const int hf    = lane >> 4;       // 0: K0/K1 half, 1: K2/K3 half
    const int mbase = blockIdx.x * MTILE;

    // ---- stage feats tile [64 x 224] into LDS (concat on the fly) ----
    for (int idx = tid; idx < MTILE * IN_DIM; idx += 256) {
        int r = idx / IN_DIM;
        int c = idx - r * IN_DIM;
        int node = mbase + r;
        if (node >= n_nodes) node = n_nodes - 1;   // clamp; stores are guarded later
        float v;
        if (c < 32)        v = rec_agg[node * E_DIM + c];
        else if (c < 64)   v = send_agg[node * E_DIM + (c - 32)];
        else if (c < 192)  v = node_attrs[node * NODE_DIM + (c - 64)];
        else               v = gattr[c - 192];
        smem[r * LDS_STRIDE + c] = v;
    }
    __syncthreads();

    // ---- GEMM1: feats[64x224] @ W1[224x256], waves split hidden cols (32 each) ----
    const int hn0 = wave * 32;
    const v8f vzero = {0.f, 0.f, 0.f, 0.f, 0.f, 0.f, 0.f, 0.f};
    v8f acc[4][2];
    #pragma unroll
    for (int mt = 0; mt < 4; ++mt) { acc[mt][0] = vzero; acc[mt][1] = vzero; }

    for (int k0 = 0; k0 < IN_DIM; k0 += 4) {
        const int krow = k0 + 2 * hf;
        // B frags (4x16 f32): lane half hf covers K rows krow, krow+1
        v2f bA, bB;
        bA.x = W1[krow * HID + hn0 + ln];
        bA.y = W1[(krow + 1) * HID + hn0 + ln];
        bB.x = W1[krow * HID + hn0 + 16 + ln];
        bB.y = W1[(krow + 1) * HID + hn0 + 16 + ln];
        #pragma unroll
        for (int mt = 0; mt < 4; ++mt) {
            v2f a = *(const v2f*)&smem[(mt * 16 + ln) * LDS_STRIDE + krow];
            acc[mt][0] = __builtin_amdgcn_wmma_f32_16x16x4_f32(
                false, a, false, bA, (short)0, acc[mt][0], false, false);
            acc[mt][1] = __builtin_amdgcn_wmma_f32_16x16x4_f32(
                false, a, false, bB, (short)0, acc[mt][1], false, false);
        }
    }
    __syncthreads();   // all LDS feats reads done before overwrite

    // ---- bias + ReLU, write h[64x256] back into the same LDS buffer ----
    const float biasA = b1[hn0 + ln];
    const float biasB = b1[hn0 + 16 + ln];
    #pragma unroll
    for (int mt = 0; mt < 4; ++mt) {
        #pragma unroll
        for (int g = 0; g < 8; ++g) {
            int row = mt * 16 + g + 8 * hf;     // C/D layout: VGPR g -> M = g (+8 for hi lanes)
            float h0 = acc[mt][0][g] + biasA; h0 = h0 > 0.f ? h0 : 0.f;
            float h1 = acc[mt][1][g] + biasB; h1 = h1 > 0.f ? h1 : 0.f;
            smem[row * LDS_STRIDE + hn0 + ln]      = h0;
            smem[row * LDS_STRIDE + hn0 + 16 + ln] = h1;
        }
    }
    __syncthreads();

    // ---- GEMM2: h[64x256] @ W2[256x128], waves split output cols (16 each) ----
    const int on0 = wave * 16;
    v8f acc2[4];
    #pragma unroll
    for (int mt = 0; mt < 4; ++mt) acc2[mt] = vzero;

    for (int k0 = 0; k0 < HID; k0 += 4) {
        const int krow = k0 + 2 * hf;
        v2f b;
        b.x = W2[krow * NODE_DIM + on0 + ln];
        b.y = W2[(krow + 1) * NODE_DIM + on0 + ln];
        #pragma unroll
        for (int mt = 0; mt < 4; ++mt) {
            v2f a = *(const v2f*)&smem[(mt * 16 + ln) * LDS_STRIDE + krow];
            acc2[mt] = __builtin_amdgcn_wmma_f32_16x16x4_f32(
                false, a, false, b, (short)0, acc2[mt], false, false);
        }
    }

    // ---- bias + guarded store ----
    const float bias2 = b2[on0 + ln];
    #pragma unroll
    for (int mt = 0; mt < 4; ++mt) {
        #pragma unroll
        for (int g = 0; g < 8; ++g) {
            int row = mbase + mt * 16 + g + 8 * hf;
            if (row < n_nodes) out[row * NODE_DIM + on0 + ln] = acc2[mt][g] + bias2;
        }
    }
}

extern "C" void kernel_launch(void* const* d_in, const int* in_sizes, int n_in,
                              void* d_out, int out_size, void* d_ws, size_t ws_size,
                              hipStream_t stream) {
    const float* edge_attrs = (const float*)d_in[0];
    const int*   senders    = (const int*)d_in[1];
    const int*   receivers  = (const int*)d_in[2];
    const float* node_attrs = (const float*)d_in[3];
    const float* gattr      = (const float*)d_in[4];
    const float* W1         = (const float*)d_in[5];
    const float* b1         = (const float*)d_in[6];
    const float* W2         = (const float*)d_in[7];
    const float* b2         = (const float*)d_in[8];
    float* out = (float*)d_out;

    const int n_edges = in_sizes[1];
    const int n_nodes = in_sizes[3] / NODE_DIM;

    float* rec_agg  = (float*)d_ws;
    float* send_agg = rec_agg + (size_t)n_nodes * E_DIM;

    // 1) zero aggregates
    int nz = 2 * n_nodes * E_DIM;
    zero_ws_kernel<<<(nz + 255) / 256, 256, 0, stream>>>(rec_agg, nz);

    // 2) fused double segment-sum scatter (one wave per edge)
    edge_scatter_kernel<<<(n_edges + 7) / 8, 256, 0, stream>>>(
        edge_attrs, senders, receivers, rec_agg, send_agg, n_edges);

    // 3) fused concat + MLP (fp32 WMMA)
    node_mlp_kernel<<<(n_nodes + MTILE - 1) / MTILE, 256, 0, stream>>>(
        rec_agg, send_agg, node_attrs, gattr, W1, b1, W2, b2, out, n_nodes);
}